// TemporalProcessor_89558658056733
// MI455X (gfx1250) — compile-verified
//
#include <hip/hip_runtime.h>
#include <hip/hip_bf16.h>
#include <math.h>

typedef __attribute__((ext_vector_type(2))) float v2f;
typedef __attribute__((ext_vector_type(8))) float v8f;

#define TOKENS   16384      // 8 * 2048
#define SEQ      2048
#define BATCH    8
#define HID      512
#define NSTATE   256
#define FFNDIM   2048

// ---------------------------------------------------------------------------
// WMMA f32 GEMM: Y = act(X[M,K] @ W[N,K]^T + bias[N] + add[M,N])
// block = 256 threads = 8 waves (2 x 4); each wave computes a 32x32 region
// (2 M-tiles x 2 N-tiles) -> block tile 64(M) x 128(N). 2x operand reuse.
// A-frag (16x4 f32): lane m=l%16, VGPR j holds K = 4*kk + 2*(l/16) + j.
// B-frag mirrored:   lane n=l%16, B[k][n] = W[n][k].
// C/D: VGPR r -> (m = r + 8*(l/16), n = l%16).
// ATM: gather A rows from time-major (t,b,n) buffer (m = b*SEQ + s).
// OTM: scatter output rows to time-major (t,b,n) buffer.
// ---------------------------------------------------------------------------
template<int GELU, int ATM, int OTM>
__global__ __launch_bounds__(256)
void gemm_wmma_f32(const float* __restrict__ X, const float* __restrict__ W,
                   const float* __restrict__ bias, const float* __restrict__ add,
                   float* __restrict__ Y, int M, int N, int K)
{
    const int lane = threadIdx.x & 31;
    const int wave = threadIdx.x >> 5;
    const int wm = wave >> 2;                 // 0..1
    const int wn = wave & 3;                  // 0..3
    const int m0 = blockIdx.y * 64 + wm * 32;
    const int n0 = blockIdx.x * 128 + wn * 32;
    const int lm = lane & 15;
    const int lh = lane >> 4;

    const int mr0 = m0 + lm, mr1 = m0 + 16 + lm;
    const size_t xr0 = ATM ? ((size_t)(mr0 & (SEQ - 1)) * BATCH + (mr0 >> 11))
                           : (size_t)mr0;
    const size_t xr1 = ATM ? ((size_t)(mr1 & (SEQ - 1)) * BATCH + (mr1 >> 11))
                           : (size_t)mr1;
    const float* __restrict__ xrow0 = X + xr0 * K;
    const float* __restrict__ xrow1 = X + xr1 * K;
    const float* __restrict__ wrow0 = W + (size_t)(n0 + lm) * K;
    const float* __restrict__ wrow1 = W + (size_t)(n0 + 16 + lm) * K;

    v8f acc00 = {}, acc01 = {}, acc10 = {}, acc11 = {};
    for (int k = 0; k < K; k += 4) {
        const int kb = k + 2 * lh;
        v2f a0; a0.x = xrow0[kb]; a0.y = xrow0[kb + 1];
        v2f a1; a1.x = xrow1[kb]; a1.y = xrow1[kb + 1];
        v2f b0; b0.x = wrow0[kb]; b0.y = wrow0[kb + 1];
        v2f b1; b1.x = wrow1[kb]; b1.y = wrow1[kb + 1];
        acc00 = __builtin_amdgcn_wmma_f32_16x16x4_f32(false, a0, false, b0, (short)0, acc00, false, false);
        acc01 = __builtin_amdgcn_wmma_f32_16x16x4_f32(false, a0, false, b1, (short)0, acc01, false, false);
        acc10 = __builtin_amdgcn_wmma_f32_16x16x4_f32(false, a1, false, b0, (short)0, acc10, false, false);
        acc11 = __builtin_amdgcn_wmma_f32_16x16x4_f32(false, a1, false, b1, (short)0, acc11, false, false);
    }

#pragma unroll
    for (int i = 0; i < 2; ++i) {
#pragma unroll
        for (int j = 0; j < 2; ++j) {
            const v8f& acc = i ? (j ? acc11 : acc10) : (j ? acc01 : acc00);
#pragma unroll
            for (int r = 0; r < 8; ++r) {
                const int m = m0 + i * 16 + r + 8 * lh;
                const int n = n0 + j * 16 + lm;
                float v = acc[r];
                if (bias) v += bias[n];
                if (add)  v += add[(size_t)m * N + n];
                if (GELU) v = 0.5f * v * (1.0f + erff(v * 0.70710678118654752440f));
                const size_t om = OTM ? ((size_t)(m & (SEQ - 1)) * BATCH + (m >> 11))
                                      : (size_t)m;
                Y[om * N + n] = v;
            }
        }
    }
}

// ---------------------------------------------------------------------------
// RMSNorm over last dim (H=512): one block per row.
// ---------------------------------------------------------------------------
__global__ __launch_bounds__(256)
void rmsnorm_kernel(const float* __restrict__ X, const float* __restrict__ w,
                    float* __restrict__ Y, int H)
{
    __shared__ float red[8];
    const size_t row = blockIdx.x;
    const float* __restrict__ x = X + row * H;
    float ss = 0.f;
    for (int i = threadIdx.x; i < H; i += 256) { float v = x[i]; ss += v * v; }
#pragma unroll
    for (int off = 16; off > 0; off >>= 1) ss += __shfl_down(ss, off, 32);
    if ((threadIdx.x & 31) == 0) red[threadIdx.x >> 5] = ss;
    __syncthreads();
    if (threadIdx.x == 0) {
        float t = 0.f;
#pragma unroll
        for (int i = 0; i < 8; ++i) t += red[i];
        red[0] = rsqrtf(t / (float)H + 1e-6f);
    }
    __syncthreads();
    const float scale = red[0];
    for (int i = threadIdx.x; i < H; i += 256)
        Y[row * H + i] = x[i] * scale * w[i];
}

// ---------------------------------------------------------------------------
// Discretize: A_d = I - exp(A_log)*dt  (256x256), B_d = B*dt (256x512)
// ---------------------------------------------------------------------------
__global__ __launch_bounds__(256)
void discretize_kernel(const float* __restrict__ A_log, const float* __restrict__ B,
                       const float* __restrict__ dt_log,
                       float* __restrict__ A_d, float* __restrict__ B_d)
{
    const float dt = expf(dt_log[0]);
    const int i = blockIdx.x * 256 + threadIdx.x;      // 0 .. 256*512-1
    if (i < NSTATE * NSTATE) {
        const int r = i >> 8, c = i & 255;
        A_d[i] = (r == c ? 1.0f : 0.0f) - expf(A_log[i]) * dt;
    }
    if (i < NSTATE * HID) B_d[i] = B[i] * dt;
}

// ---------------------------------------------------------------------------
// SSM scan: s_{t+1} = s_t @ A_d^T + Bu_t, all 8 batches packed into WMMA M dim.
// One block, 16 waves; wave j owns N-columns [16j,16j+16). A_d^T tile fragments
// stay resident in VGPRs across all 2048 steps; s double-buffered in LDS.
// Bu/states are TIME-MAJOR: idx = (t*8 + b)*256 + n -> one contiguous 8KB row
// of global traffic per step on the sequential critical path.
// ---------------------------------------------------------------------------
#define SPAD 260
__global__ __launch_bounds__(512)
void ssm_scan_kernel(const float* __restrict__ Bu, const float* __restrict__ A_d,
                     float* __restrict__ states, float* __restrict__ finals)
{
    __shared__ float sbuf[2][16 * SPAD];
    const int lane = threadIdx.x & 31;
    const int wave = threadIdx.x >> 5;   // N-tile index 0..15
    const int n0 = wave * 16;
    const int lm = lane & 15;
    const int lh = lane >> 4;

    // Stationary B fragments: B[k][n] = A_d^T[k][n0+n] = A_d[n0+n][k]
    v2f bfrag[64];
    {
        const float* __restrict__ arow = A_d + (size_t)(n0 + lm) * NSTATE;
#pragma unroll
        for (int kk = 0; kk < 64; ++kk) {
            const int kb = kk * 4 + 2 * lh;
            bfrag[kk].x = arow[kb];
            bfrag[kk].y = arow[kb + 1];
        }
    }
    for (int i = threadIdx.x; i < 16 * SPAD; i += 512) {
        sbuf[0][i] = 0.f; sbuf[1][i] = 0.f;
    }
    __syncthreads();

    v8f acc = {};
    int cur = 0;
    for (int t = 0; t < SEQ; ++t) {
        const float* __restrict__ burow = Bu + (size_t)t * (BATCH * NSTATE);
        float* __restrict__ strow = states + (size_t)t * (BATCH * NSTATE);
        // Seed accumulator with Bu_t (rows = batches 0..7; rows 8..15 zero)
#pragma unroll
        for (int r = 0; r < 8; ++r) {
            const int m = r + 8 * lh;           // batch row in 16-row tile
            float v = 0.f;
            if (m < BATCH) v = burow[m * NSTATE + n0 + lm];
            acc[r] = v;
        }
        const float* __restrict__ s = sbuf[cur];
#pragma unroll
        for (int kk = 0; kk < 64; ++kk) {
            const int kb = kk * 4 + 2 * lh;
            v2f a; a.x = s[lm * SPAD + kb]; a.y = s[lm * SPAD + kb + 1];
            acc = __builtin_amdgcn_wmma_f32_16x16x4_f32(false, a, false, bfrag[kk],
                                                        (short)0, acc, false, false);
        }
        float* __restrict__ sn = sbuf[cur ^ 1];
#pragma unroll
        for (int r = 0; r < 8; ++r) {
            const int m = r + 8 * lh;
            sn[m * SPAD + n0 + lm] = acc[r];
            if (m < BATCH)
                strow[m * NSTATE + n0 + lm] = acc[r];
        }
        cur ^= 1;
        __syncthreads();
    }
#pragma unroll
    for (int r = 0; r < 8; ++r) {
        const int m = r + 8 * lh;
        if (m < BATCH) finals[(size_t)m * NSTATE + n0 + lm] = acc[r];
    }
}

// ---------------------------------------------------------------------------
// a += b
// ---------------------------------------------------------------------------
__global__ __launch_bounds__(256)
void add_kernel(float* __restrict__ a, const float* __restrict__ b, int n)
{
    const int i = blockIdx.x * 256 + threadIdx.x;
    if (i < n) a[i] += b[i];
}

// ---------------------------------------------------------------------------
// Host side
// ---------------------------------------------------------------------------
struct LayerP {
    const float *A_log, *B, *C, *D, *dt_log, *norm_w, *norm2_w,
                *ffn_w1, *ffn_b1, *ffn_w2, *ffn_b2;
};

extern "C" void kernel_launch(void* const* d_in, const int* in_sizes, int n_in,
                              void* d_out, int out_size, void* d_ws, size_t ws_size,
                              hipStream_t stream)
{
    (void)out_size; (void)ws_size;
    // ---- Input order detection (insertion vs jax-sorted; x first vs last) ----
    const float* x;
    int p0;                       // index of first params entry
    if (in_sizes[0] == TOKENS * HID && n_in > 1 && in_sizes[1] != TOKENS * HID) {
        x = (const float*)d_in[0]; p0 = 1;           // x first
    } else {
        x = (const float*)d_in[n_in - 1]; p0 = 0;    // params first (sorted pytree)
    }
    const bool sorted = (in_sizes[p0] == HID);       // in_b(512) first => sorted keys

    const float *in_w, *in_b, *out_w, *out_b;
    LayerP L[4];
    if (!sorted) {
        in_w = (const float*)d_in[p0 + 0];
        in_b = (const float*)d_in[p0 + 1];
        for (int l = 0; l < 4; ++l) {
            const int b = p0 + 2 + 11 * l;
            L[l].A_log  = (const float*)d_in[b + 0];
            L[l].B      = (const float*)d_in[b + 1];
            L[l].C      = (const float*)d_in[b + 2];
            L[l].D      = (const float*)d_in[b + 3];
            L[l].dt_log = (const float*)d_in[b + 4];
            L[l].norm_w = (const float*)d_in[b + 5];
            L[l].norm2_w= (const float*)d_in[b + 6];
            L[l].ffn_w1 = (const float*)d_in[b + 7];
            L[l].ffn_b1 = (const float*)d_in[b + 8];
            L[l].ffn_w2 = (const float*)d_in[b + 9];
            L[l].ffn_b2 = (const float*)d_in[b + 10];
        }
        out_w = (const float*)d_in[p0 + 46];
        out_b = (const float*)d_in[p0 + 47];
    } else {
        in_b = (const float*)d_in[p0 + 0];
        in_w = (const float*)d_in[p0 + 1];
        for (int l = 0; l < 4; ++l) {
            const int b = p0 + 2 + 11 * l;   // sorted: A_log,B,C,D,dt_log,ffn_b1,ffn_b2,ffn_w1,ffn_w2,norm2_w,norm_w
            L[l].A_log  = (const float*)d_in[b + 0];
            L[l].B      = (const float*)d_in[b + 1];
            L[l].C      = (const float*)d_in[b + 2];
            L[l].D      = (const float*)d_in[b + 3];
            L[l].dt_log = (const float*)d_in[b + 4];
            L[l].ffn_b1 = (const float*)d_in[b + 5];
            L[l].ffn_b2 = (const float*)d_in[b + 6];
            L[l].ffn_w1 = (const float*)d_in[b + 7];
            L[l].ffn_w2 = (const float*)d_in[b + 8];
            L[l].norm2_w= (const float*)d_in[b + 9];
            L[l].norm_w = (const float*)d_in[b + 10];
        }
        out_b = (const float*)d_in[p0 + 46];
        out_w = (const float*)d_in[p0 + 47];
    }

    // ---- Workspace carving (floats) ----
    float* Wp  = (float*)d_ws;
    float* h   = Wp;                        // 16384*512
    float* u   = h   + (size_t)TOKENS * HID;
    float* t1  = u   + (size_t)TOKENS * HID;
    float* f   = t1  + (size_t)TOKENS * HID;       // 16384*2048
    float* Bu  = f   + (size_t)TOKENS * FFNDIM;    // 16384*256 (time-major)
    float* st  = Bu  + (size_t)TOKENS * NSTATE;    // 16384*256 (time-major)
    float* A_d = st  + (size_t)TOKENS * NSTATE;    // 256*256
    float* B_d = A_d + (size_t)NSTATE * NSTATE;    // 256*512

    float* out    = (float*)d_out;
    float* finals = out + (size_t)TOKENS * HID;

    const dim3 blk(256);
    const dim3 gN512 (HID    / 128, TOKENS / 64);
    const dim3 gN256 (NSTATE / 128, TOKENS / 64);
    const dim3 gN2048(FFNDIM / 128, TOKENS / 64);
    const int nElems = TOKENS * HID;

    // h = x @ in_w^T + in_b
    gemm_wmma_f32<0,0,0><<<gN512, blk, 0, stream>>>(x, in_w, in_b, nullptr, h,
                                                    TOKENS, HID, HID);
    for (int l = 0; l < 4; ++l) {
        const LayerP& p = L[l];
        discretize_kernel<<<(NSTATE * HID) / 256, blk, 0, stream>>>(
            p.A_log, p.B, p.dt_log, A_d, B_d);
        // u = rmsnorm(h)
        rmsnorm_kernel<<<TOKENS, blk, 0, stream>>>(h, p.norm_w, u, HID);
        // Bu = u @ B_d^T   (scattered to time-major for the scan)
        gemm_wmma_f32<0,0,1><<<gN256, blk, 0, stream>>>(u, B_d, nullptr, nullptr, Bu,
                                                        TOKENS, NSTATE, HID);
        // sequential scan -> states (time-major), finals
        ssm_scan_kernel<<<1, 512, 0, stream>>>(Bu, A_d, st, finals + l * BATCH * NSTATE);
        // t1 = states @ C^T   (gather A rows from time-major states)
        gemm_wmma_f32<0,1,0><<<gN512, blk, 0, stream>>>(st, p.C, nullptr, nullptr, t1,
                                                        TOKENS, HID, NSTATE);
        // t1 = u @ D^T + t1   (y)
        gemm_wmma_f32<0,0,0><<<gN512, blk, 0, stream>>>(u, p.D, nullptr, t1, t1,
                                                        TOKENS, HID, HID);
        // h += y
        add_kernel<<<nElems / 256, blk, 0, stream>>>(h, t1, nElems);
        // u = rmsnorm(h) (norm2)
        rmsnorm_kernel<<<TOKENS, blk, 0, stream>>>(h, p.norm2_w, u, HID);
        // f = gelu(u @ ffn_w1^T + b1)
        gemm_wmma_f32<1,0,0><<<gN2048, blk, 0, stream>>>(u, p.ffn_w1, p.ffn_b1, nullptr, f,
                                                         TOKENS, FFNDIM, HID);
        // t1 = f @ ffn_w2^T + b2
        gemm_wmma_f32<0,0,0><<<gN512, blk, 0, stream>>>(f, p.ffn_w2, p.ffn_b2, nullptr, t1,
                                                        TOKENS, HID, FFNDIM);
        // h += t1
        add_kernel<<<nElems / 256, blk, 0, stream>>>(h, t1, nElems);
    }
    // out = h @ out_w^T + out_b
    gemm_wmma_f32<0,0,0><<<gN512, blk, 0, stream>>>(h, out_w, out_b, nullptr, out,
                                                    TOKENS, HID, HID);
}